// CTCLayer_34187939676295
// MI455X (gfx1250) — compile-verified
//
#include <hip/hip_runtime.h>
#include <stdint.h>

// ---- problem constants (match reference) ----
#define B_    256
#define T_    512
#define V_    256
#define L_    64
#define S_    129          // 2*L + 1
#define BLANK 255          // V-1
#define EPSF  1e-7f
#define NEGF  (-1e30f)

#define DEPTH 16           // async prefetch depth in rows (2 async instrs per row)
#define KS    5            // states per lane: ceil(129/32)

typedef int v4i __attribute__((ext_vector_type(4)));
typedef __attribute__((address_space(3))) v4i lds_v4i;   // LDS-qualified vector
typedef __attribute__((address_space(3))) void LAS;      // LDS void (for asm path)

#ifndef __has_builtin
#define __has_builtin(x) 0
#endif
#if __has_builtin(__builtin_amdgcn_global_load_async_to_lds_b128)
#define HAVE_ASYNC_B 1
#else
#define HAVE_ASYNC_B 0
#endif
#if __has_builtin(__builtin_amdgcn_s_wait_asynccnt)
#define HAVE_WAIT_B 1
#else
#define HAVE_WAIT_B 0
#endif

// One lane moves 16B global -> LDS, async (tracked by ASYNCcnt).
template <int OFF>
__device__ __forceinline__ void async_b128(const float* g, float* l) {
#if HAVE_ASYNC_B
  __builtin_amdgcn_global_load_async_to_lds_b128((v4i*)g, (lds_v4i*)l, OFF, 0);
#else
  uint32_t lo = (uint32_t)(uintptr_t)(LAS*)l;      // LDS byte offset
  uint64_t ga = (uint64_t)(uintptr_t)g;
  if (OFF == 0)
    asm volatile("global_load_async_to_lds_b128 %0, %1, off"
                 :: "v"(lo), "v"(ga) : "memory");
  else
    asm volatile("global_load_async_to_lds_b128 %0, %1, off offset:512"
                 :: "v"(lo), "v"(ga) : "memory");
#endif
}

template <int N>
__device__ __forceinline__ void wait_async() {
#if HAVE_WAIT_B
  __builtin_amdgcn_s_wait_asynccnt((unsigned short)N);
#else
  asm volatile("s_wait_asynccnt %0" :: "n"(N) : "memory");
#endif
  asm volatile("" ::: "memory");   // compiler ordering fence for LDS reads
}

// logaddexp3 in f32: mx + log(sum exp(.-mx)); NEG terms vanish (exp underflow -> 0).
__device__ __forceinline__ float lae3(float a, float b, float c) {
  float mx  = fmaxf(fmaxf(a, b), c);
  float sum = __expf(a - mx) + __expf(b - mx) + __expf(c - mx);
  return mx + __logf(sum);
}

__global__ __launch_bounds__(32)
void ctc_batch_cost_kernel(const int* __restrict__ yt,     // [B, L] int32
                           const float* __restrict__ yp,   // [B, T, V] f32 probs
                           float* __restrict__ out) {      // [B] f32 loss
  __shared__ float rowbuf[DEPTH][V_];    // 16 KB: cyclic async staging buffer

  const int lane = threadIdx.x & 31;
  const int b    = blockIdx.x;           // one wave32 block per batch element

  const float* ypb = yp + (size_t)b * T_ * V_;
  const int*   ytb = yt + b * L_;

  // ---- per-state metadata (fixed over t): label index, skip flag, validity ----
  int  lab[KS];
  bool skp[KS];
  bool val[KS];
#pragma unroll
  for (int k = 0; k < KS; ++k) {
    int s  = lane + 32 * k;
    val[k] = (s < S_);
    int sc = val[k] ? s : 0;
    if (sc & 1) {
      int i    = sc >> 1;
      int cur  = ytb[i];
      lab[k]   = cur;
      int pm2  = (i > 0) ? ytb[i - 1] : -1;
      skp[k]   = (cur != pm2);          // odd state: skip iff label differs from s-2
    } else {
      lab[k] = BLANK;                   // even state: blank, never skip
      skp[k] = false;
    }
  }

  // ---- prologue: async-prefetch rows 0 .. DEPTH-2 ----
#pragma unroll
  for (int r = 0; r < DEPTH - 1; ++r) {
    const float* g = ypb + (size_t)r * V_ + lane * 4;
    float*       l = &rowbuf[r][lane * 4];
    async_b128<0>(g, l);
    async_b128<512>(g, l);
  }

  float a[KS];
#pragma unroll
  for (int k = 0; k < KS; ++k) a[k] = NEGF;

  // ---- time scan: issue row t+D-1, wait row t, DP step ----
  for (int t = 0; t < T_; ++t) {
    {
      int r = t + DEPTH - 1;
      if (r > T_ - 1) r = T_ - 1;               // clamped tail re-loads hit dead slots
      int slot = (t + DEPTH - 1) % DEPTH;
      const float* g = ypb + (size_t)r * V_ + lane * 4;
      float*       l = &rowbuf[slot][lane * 4];
      async_b128<0>(g, l);
      async_b128<512>(g, l);
    }
    wait_async<2 * (DEPTH - 1)>();              // oldest (row t) now resident in LDS

    const float* row = &rowbuf[t % DEPTH][0];
    float lpv[KS];
#pragma unroll
    for (int k = 0; k < KS; ++k) lpv[k] = __logf(row[lab[k]] + EPSF);

    if (t == 0) {
#pragma unroll
      for (int k = 0; k < KS; ++k) {
        int s = lane + 32 * k;
        a[k]  = (val[k] && s < 2) ? lpv[k] : NEGF;   // only s=0,1 reachable at t=0
      }
    } else {
      float an[KS];
#pragma unroll
      for (int k = 0; k < KS; ++k) {
        float up1  = __shfl_up(a[k], 1, 32);
        float up2  = __shfl_up(a[k], 2, 32);
        float prev = (k > 0) ? a[k - 1] : NEGF;      // register-boundary neighbors
        float p31  = __shfl(prev, 31, 32);
        float p30  = __shfl(prev, 30, 32);
        float a1   = (lane == 0) ? p31 : up1;                         // alpha[s-1]
        float a2   = (lane == 0) ? p30 : ((lane == 1) ? p31 : up2);   // alpha[s-2]
        float c    = skp[k] ? a2 : NEGF;
        float m    = lae3(a[k], a1, c);
        an[k]      = val[k] ? (m + lpv[k]) : NEGF;
      }
#pragma unroll
      for (int k = 0; k < KS; ++k) a[k] = an[k];
    }
  }

  wait_async<0>();   // drain tail prefetches before wave exit

  // loss = -logaddexp(alpha[S-1], alpha[S-2]);  s=128 -> lane0,k=4;  s=127 -> lane31,k=3
  float v127 = __shfl(a[3], 31, 32);
  float v128 = __shfl(a[4], 0, 32);
  float mx   = fmaxf(v127, v128);
  float loss = -(mx + __logf(__expf(v127 - mx) + __expf(v128 - mx)));
  if (lane == 0) out[b] = loss;
}

extern "C" void kernel_launch(void* const* d_in, const int* in_sizes, int n_in,
                              void* d_out, int out_size, void* d_ws, size_t ws_size,
                              hipStream_t stream) {
  (void)in_sizes; (void)n_in; (void)out_size; (void)d_ws; (void)ws_size;
  const int*   yt  = (const int*)d_in[0];     // y_true [B, L] int32
  const float* yp  = (const float*)d_in[1];   // y_pred [B, T, V] float32
  float*       out = (float*)d_out;           // [B, 1] float32

  dim3 grid(B_);            // 256 blocks: one wave32 per batch element, max WGP spread
  dim3 block(32);
  hipLaunchKernelGGL(ctc_batch_cost_kernel, grid, block, 0, stream, yt, yp, out);
}